// GINModel_36421322670666
// MI455X (gfx1250) — compile-verified
//
#include <hip/hip_runtime.h>
#include <hip/hip_bf16.h>

// Problem constants (match reference)
#define NN 50000
#define EE 600000
#define DD 128
#define HH 128
#define LL 4
#define GG 256
#define CC 10
#define BN_EPS 1e-5f

typedef __attribute__((ext_vector_type(2))) float v2f;
typedef __attribute__((ext_vector_type(8))) float v8f;

// ---------------------------------------------------------------------------
// float atomic add -> global_atomic_add_f32
__device__ __forceinline__ void atomAdd(float* p, float v) {
    unsafeAtomicAdd(p, v);
}

// ---------------------------------------------------------------------------
// z = (1+eps)*h  (float4 vectorized); block 0 also zeroes the 4x128 BN stat
// accumulators for this layer (256 threads x 2 entries = all 512).
__global__ __launch_bounds__(256) void init_z_kernel(
    const float* __restrict__ h, float* __restrict__ z,
    const float* __restrict__ eps, int layer, float* __restrict__ stats)
{
    if (blockIdx.x == 0) {
        stats[threadIdx.x] = 0.0f;
        stats[threadIdx.x + 256] = 0.0f;
    }
    int idx = blockIdx.x * 256 + threadIdx.x;          // N*32 float4 slots
    float ep = 1.0f + eps[layer];
    float4 v = ((const float4*)h)[idx];
    v.x *= ep; v.y *= ep; v.z *= ep; v.w *= ep;
    ((float4*)z)[idx] = v;
}

// ---------------------------------------------------------------------------
// z[dst] += h[src] over all edges.  One wave per edge, float4 per lane.
// h (25.6MB) is L2-resident on MI455X (192MB L2), so the random gather and the
// scatter atomics both stay on-chip.
__global__ __launch_bounds__(256) void edge_scatter_kernel(
    const float* __restrict__ h, const int* __restrict__ ei,
    float* __restrict__ z)
{
    int e = blockIdx.x * 8 + (threadIdx.x >> 5);
    if (e >= EE) return;
    int lane = threadIdx.x & 31;
    int src = ei[e];
    int dst = ei[EE + e];
    float4 v = *(const float4*)(h + src * DD + lane * 4);
    float* p = z + dst * DD + lane * 4;
    atomAdd(p + 0, v.x);
    atomAdd(p + 1, v.y);
    atomAdd(p + 2, v.z);
    atomAdd(p + 3, v.w);
}

// ---------------------------------------------------------------------------
// out = act(Z) @ W + bias, where act = identity (BN_IN=false) or
// x -> relu(x*scale+shift) (BN_IN=true), fused column sum/sumsq accumulation
// for the following BatchNorm.
// Block: 256 threads = 8 waves, covers 64 rows x 128 cols.
// Wave w: row-tile w>>1 (16 rows), column half w&1 (4 col tiles of 16).
// K = 128, stepped by 4 with V_WMMA_F32_16X16X4_F32 (exact fp32).
template <bool BN_IN>
__global__ __launch_bounds__(256) void gemm_kernel(
    const float* __restrict__ Z, const float* __restrict__ W,
    const float* __restrict__ bias, float* __restrict__ out,
    float* __restrict__ colsum, float* __restrict__ colsq,
    const float* __restrict__ scale, const float* __restrict__ shift)
{
    __shared__ float zs[64 * 132];                     // padded pitch: no bank conflicts
    const int tid = threadIdx.x;
    const int rowbase = blockIdx.x * 64;

    // cooperative staging of the 64x128 input tile (float4, 16B-aligned in LDS)
    for (int i = tid; i < 64 * 32; i += 256) {
        int r = i >> 5;
        int c4 = (i & 31) << 2;
        int gr = rowbase + r;
        if (gr < NN) {
            float4 v = *(const float4*)(Z + gr * DD + c4);
            if (BN_IN) {
                v.x = fmaxf(v.x * scale[c4 + 0] + shift[c4 + 0], 0.0f);
                v.y = fmaxf(v.y * scale[c4 + 1] + shift[c4 + 1], 0.0f);
                v.z = fmaxf(v.z * scale[c4 + 2] + shift[c4 + 2], 0.0f);
                v.w = fmaxf(v.w * scale[c4 + 3] + shift[c4 + 3], 0.0f);
            }
            *(float4*)(zs + r * 132 + c4) = v;
        }
    }
    __syncthreads();

    const int wave = tid >> 5;
    const int lane = tid & 31;
    const int half = lane >> 4;                        // 0 | 1
    const int n    = lane & 15;
    const int rt   = wave >> 1;                        // 0..3
    const int colbase = (wave & 1) * 64;
    const int row0 = rowbase + rt * 16;
    if (row0 >= NN) return;                            // wave-uniform (N % 16 == 0)

    v8f acc[4];
#pragma unroll
    for (int ct = 0; ct < 4; ++ct) {
        float bv = bias[colbase + ct * 16 + n];
#pragma unroll
        for (int v = 0; v < 8; ++v) acc[ct][v] = bv;
    }

    const float* arow = zs + (rt * 16 + n) * 132 + 2 * half;
#pragma unroll 4
    for (int ks = 0; ks < DD; ks += 4) {
        v2f a;
        a.x = arow[ks];
        a.y = arow[ks + 1];
#pragma unroll
        for (int ct = 0; ct < 4; ++ct) {
            const float* wp = W + (ks + 2 * half) * HH + colbase + ct * 16 + n;
            v2f b;
            b.x = wp[0];
            b.y = wp[HH];
            acc[ct] = __builtin_amdgcn_wmma_f32_16x16x4_f32(
                false, a, false, b, (short)0, acc[ct], false, false);
        }
    }

#pragma unroll
    for (int ct = 0; ct < 4; ++ct) {
        int col = colbase + ct * 16 + n;
        float s = 0.0f, q = 0.0f;
#pragma unroll
        for (int v = 0; v < 8; ++v) {
            float val = acc[ct][v];
            out[(row0 + v + 8 * half) * HH + col] = val;
            s += val;
            q += val * val;
        }
        s += __shfl_xor(s, 16, 32);                    // combine the two row-halves
        q += __shfl_xor(q, 16, 32);
        if (half == 0) {
            atomAdd(&colsum[col], s);
            atomAdd(&colsq[col], q);
        }
    }
}

// ---------------------------------------------------------------------------
// Convert column sum/sumsq into BN affine: y = x*scale + shift
__global__ void bn_stats_kernel(
    const float* __restrict__ sum, const float* __restrict__ sumsq,
    const float* __restrict__ g, const float* __restrict__ bt,
    float* __restrict__ scale, float* __restrict__ shift)
{
    int c = threadIdx.x;                               // 128 threads
    const float invN = 1.0f / (float)NN;
    float mu  = sum[c] * invN;
    float var = sumsq[c] * invN - mu * mu;
    float sc  = g[c] * rsqrtf(var + BN_EPS);
    scale[c] = sc;
    shift[c] = bt[c] - mu * sc;
}

// ---------------------------------------------------------------------------
// h = relu(BN(tmp)); pooled[batch[n], layer*H + d] += h   (graph readout)
__global__ __launch_bounds__(256) void bn_pool_kernel(
    const float* __restrict__ t, const float* __restrict__ scale,
    const float* __restrict__ shift, float* __restrict__ h,
    float* __restrict__ pooled, const int* __restrict__ batch, int layer)
{
    int idx = blockIdx.x * 256 + threadIdx.x;          // N*32 float4 slots
    int nrow = idx >> 5;
    int q = (idx & 31) << 2;
    float4 v = *(const float4*)(t + nrow * DD + q);
    v.x = fmaxf(v.x * scale[q + 0] + shift[q + 0], 0.0f);
    v.y = fmaxf(v.y * scale[q + 1] + shift[q + 1], 0.0f);
    v.z = fmaxf(v.z * scale[q + 2] + shift[q + 2], 0.0f);
    v.w = fmaxf(v.w * scale[q + 3] + shift[q + 3], 0.0f);
    *(float4*)(h + nrow * DD + q) = v;
    float* pp = pooled + batch[nrow] * (HH * LL) + layer * HH + q;
    atomAdd(pp + 0, v.x);
    atomAdd(pp + 1, v.y);
    atomAdd(pp + 2, v.z);
    atomAdd(pp + 3, v.w);
}

__global__ void zero_kernel(float* __restrict__ p, int nfloat)
{
    int i = blockIdx.x * 256 + threadIdx.x;
    if (i < nfloat) p[i] = 0.0f;
}

// ---------------------------------------------------------------------------
// fc1: relu(pooled[256,512] @ W[512,128] + b) via WMMA.
// 16 blocks x 8 waves; block = one 16-row tile, wave = one 16-col tile.
__global__ __launch_bounds__(256) void fc1_kernel(
    const float* __restrict__ pooled, const float* __restrict__ W,
    const float* __restrict__ bias, float* __restrict__ out)
{
    __shared__ float ps[16 * 516];
    const int tid = threadIdx.x;
    const int rowbase = blockIdx.x * 16;
    for (int i = tid; i < 16 * 128; i += 256) {        // 2048 float4s
        int r = i >> 7;
        int c4 = (i & 127) << 2;
        float4 v = *(const float4*)(pooled + (rowbase + r) * (HH * LL) + c4);
        *(float4*)(ps + r * 516 + c4) = v;
    }
    __syncthreads();

    const int wave = tid >> 5;
    const int lane = tid & 31;
    const int half = lane >> 4;
    const int n    = lane & 15;
    const int col  = wave * 16 + n;

    v8f acc;
    float bv = bias[col];
#pragma unroll
    for (int v = 0; v < 8; ++v) acc[v] = bv;

    const float* arow = ps + n * 516 + 2 * half;
#pragma unroll 4
    for (int ks = 0; ks < HH * LL; ks += 4) {
        v2f a;
        a.x = arow[ks];
        a.y = arow[ks + 1];
        const float* wp = W + (ks + 2 * half) * HH + col;
        v2f b;
        b.x = wp[0];
        b.y = wp[HH];
        acc = __builtin_amdgcn_wmma_f32_16x16x4_f32(
            false, a, false, b, (short)0, acc, false, false);
    }
#pragma unroll
    for (int v = 0; v < 8; ++v)
        out[(rowbase + v + 8 * half) * HH + col] = fmaxf(acc[v], 0.0f);
}

// ---------------------------------------------------------------------------
// fc2: [256,128] @ [128,10] + b  (tiny, plain VALU)
__global__ __launch_bounds__(256) void fc2_kernel(
    const float* __restrict__ a, const float* __restrict__ W,
    const float* __restrict__ bias, float* __restrict__ out)
{
    int idx = blockIdx.x * 256 + threadIdx.x;
    if (idx >= GG * CC) return;
    int g = idx / CC, c = idx % CC;
    float s = bias[c];
#pragma unroll 8
    for (int k = 0; k < HH; ++k) s += a[g * HH + k] * W[k * CC + c];
    out[idx] = s;
}

// ---------------------------------------------------------------------------
extern "C" void kernel_launch(void* const* d_in, const int* in_sizes, int n_in,
                              void* d_out, int out_size, void* d_ws, size_t ws_size,
                              hipStream_t stream)
{
    const float* x     = (const float*)d_in[0];
    const int*   ei    = (const int*)d_in[1];
    const int*   batch = (const int*)d_in[2];
    const float* W1    = (const float*)d_in[3];
    const float* b1    = (const float*)d_in[4];
    const float* g1    = (const float*)d_in[5];
    const float* bt1   = (const float*)d_in[6];
    const float* W2    = (const float*)d_in[7];
    const float* b2    = (const float*)d_in[8];
    const float* g2    = (const float*)d_in[9];
    const float* bt2   = (const float*)d_in[10];
    const float* eps   = (const float*)d_in[11];
    const float* fc1w  = (const float*)d_in[12];
    const float* fc1b  = (const float*)d_in[13];
    const float* fc2w  = (const float*)d_in[14];
    const float* fc2b  = (const float*)d_in[15];

    // workspace layout (floats)
    float* ws     = (float*)d_ws;
    float* zbuf   = ws;                       // N*D
    float* tmp1   = zbuf  + (size_t)NN * DD;  // N*H
    float* tmp2   = tmp1  + (size_t)NN * HH;  // N*H
    float* hbuf   = tmp2  + (size_t)NN * HH;  // N*H
    float* pooled = hbuf  + (size_t)NN * HH;  // G*H*L
    float* fc1out = pooled + (size_t)GG * HH * LL;  // G*H
    float* stats  = fc1out + (size_t)GG * HH;       // 4*128: sum1,sq1,sum2,sq2
    float* bnp    = stats + 512;                    // 4*128: scale1,shift1,scale2,shift2
    float* sum1 = stats, *sq1 = stats + 128, *sum2 = stats + 256, *sq2 = stats + 384;
    float* scale1 = bnp, *shift1 = bnp + 128, *scale2 = bnp + 256, *shift2 = bnp + 384;

    const int gridND   = (NN * 32) / 256;     // 6250 (N*D float4 slots)
    const int gridEdge = EE / 8;              // 75000
    const int gridGemm = (NN / 16 + 3) / 4;   // 782 (64-row blocks)

    zero_kernel<<<(GG * HH * LL + 255) / 256, 256, 0, stream>>>(pooled, GG * HH * LL);

    const float* h = x;
    for (int i = 0; i < LL; ++i) {
        init_z_kernel<<<gridND, 256, 0, stream>>>(h, zbuf, eps, i, stats);
        edge_scatter_kernel<<<gridEdge, 256, 0, stream>>>(h, ei, zbuf);
        gemm_kernel<false><<<gridGemm, 256, 0, stream>>>(
            zbuf, W1 + (size_t)i * DD * HH, b1 + i * HH, tmp1, sum1, sq1, nullptr, nullptr);
        bn_stats_kernel<<<1, 128, 0, stream>>>(sum1, sq1, g1 + i * HH, bt1 + i * HH, scale1, shift1);
        gemm_kernel<true><<<gridGemm, 256, 0, stream>>>(
            tmp1, W2 + (size_t)i * HH * HH, b2 + i * HH, tmp2, sum2, sq2, scale1, shift1);
        bn_stats_kernel<<<1, 128, 0, stream>>>(sum2, sq2, g2 + i * HH, bt2 + i * HH, scale2, shift2);
        bn_pool_kernel<<<gridND, 256, 0, stream>>>(tmp2, scale2, shift2, hbuf, pooled, batch, i);
        h = hbuf;
    }

    fc1_kernel<<<GG / 16, 256, 0, stream>>>(pooled, fc1w, fc1b, fc1out);
    fc2_kernel<<<(GG * CC + 255) / 256, 256, 0, stream>>>(fc1out, fc2w, fc2b, (float*)d_out);
}